// PrefixPEFTModel_73409581023291
// MI455X (gfx1250) — compile-verified
//
#include <hip/hip_runtime.h>
#include <cstddef>

// Problem constants from the reference
#define BB 8
#define SS 512
#define HH 16
#define VV 50265
#define MASK_ID 50264
#define LEFT_ID 4832
#define RIGHT_ID 2

typedef float v2f __attribute__((ext_vector_type(2)));
typedef float v8f __attribute__((ext_vector_type(8)));

// ---------------------------------------------------------------------------
// Kernel 1: find first occurrence of MASK/LEFT/RIGHT tokens per batch row
// (jnp.argmax(ids == TOK) semantics: first match, 0 if none).
// ---------------------------------------------------------------------------
__global__ void k_positions(const int* __restrict__ ids, int* __restrict__ posw) {
    int b = threadIdx.x;
    if (b >= BB) return;
    int mp = 0, lf = 0, rt = 0;
    bool fm = false, fl = false, fr = false;
    for (int s = 0; s < SS; ++s) {
        int v = ids[b * SS + s];
        if (!fm && v == MASK_ID)  { mp = s; fm = true; }
        if (!fl && v == LEFT_ID)  { lf = s; fl = true; }
        if (!fr && v == RIGHT_ID) { rt = s; fr = true; }
    }
    posw[b] = mp; posw[BB + b] = lf; posw[2 * BB + b] = rt;
}

// ---------------------------------------------------------------------------
// Kernel 2: log_softmax over V at the mask row; produce out_logits[b] = {neg,pos}
// One block per batch row; two-pass max / sum-exp reduction over 50265 floats.
// ---------------------------------------------------------------------------
__global__ __launch_bounds__(256)
void k_masklogits(const float* __restrict__ logits,
                  const float* __restrict__ pos_w, const float* __restrict__ neg_w,
                  const int* __restrict__ posw, float* __restrict__ out) {
    __shared__ float red[256];
    const int b = blockIdx.x;
    const int t = threadIdx.x;
    const float* row = logits + ((size_t)b * SS + posw[b]) * (size_t)VV;

    float m = -1e30f;
    for (int i = t; i < VV; i += 256) m = fmaxf(m, row[i]);
    red[t] = m; __syncthreads();
    for (int s = 128; s > 0; s >>= 1) { if (t < s) red[t] = fmaxf(red[t], red[t + s]); __syncthreads(); }
    const float mx = red[0]; __syncthreads();

    float sum = 0.f;
    for (int i = t; i < VV; i += 256) sum += expf(row[i] - mx);
    red[t] = sum; __syncthreads();
    for (int s = 128; s > 0; s >>= 1) { if (t < s) red[t] += red[t + s]; __syncthreads(); }

    if (t == 0) {
        const float lse = mx + logf(red[0]);
        const int PID[3] = {1313, 372, 205};
        const int NID[3] = {2362, 699, 103};
        // softmax over the 3 mixing weights
        float pm = fmaxf(fmaxf(pos_w[0], pos_w[1]), pos_w[2]);
        float nm = fmaxf(fmaxf(neg_w[0], neg_w[1]), neg_w[2]);
        float pe[3], ne[3], ps = 0.f, ns = 0.f;
        for (int i = 0; i < 3; ++i) { pe[i] = expf(pos_w[i] - pm); ps += pe[i]; }
        for (int i = 0; i < 3; ++i) { ne[i] = expf(neg_w[i] - nm); ns += ne[i]; }
        float pos = 0.f, neg = 0.f;
        for (int i = 0; i < 3; ++i) pos += (row[PID[i]] - lse) * (pe[i] / ps);
        for (int i = 0; i < 3; ++i) neg += (row[NID[i]] - lse) * (ne[i] / ns);
        out[b * 2 + 0] = neg;   // column 0 = neg
        out[b * 2 + 1] = pos;   // column 1 = pos
    }
}

// ---------------------------------------------------------------------------
// Kernel 3 (bandwidth-dominant, ~402 MB stream): per-map mean & inv_std over
// each [512,512] attention map. 384 blocks, float4 (b128) loads, 4-way split
// accumulators to keep float rounding error tiny, deterministic LDS tree.
// ---------------------------------------------------------------------------
__global__ __launch_bounds__(256)
void k_stats(const float* __restrict__ attn,
             float* __restrict__ mean, float* __restrict__ invstd) {
    __shared__ float rs[256], rq[256];
    const int map = blockIdx.x;                  // (((l*8)+b)*16+h), 0..383
    const int t = threadIdx.x;
    const float4* p = (const float4*)(attn + (size_t)map * (SS * SS));
    float s0 = 0.f, s1 = 0.f, s2 = 0.f, s3 = 0.f;
    float q0 = 0.f, q1 = 0.f, q2 = 0.f, q3 = 0.f;
    for (int i = t; i < (SS * SS) / 4; i += 256) {
        float4 v = p[i];
        s0 += v.x; s1 += v.y; s2 += v.z; s3 += v.w;
        q0 = fmaf(v.x, v.x, q0); q1 = fmaf(v.y, v.y, q1);
        q2 = fmaf(v.z, v.z, q2); q3 = fmaf(v.w, v.w, q3);
    }
    rs[t] = (s0 + s1) + (s2 + s3);
    rq[t] = (q0 + q1) + (q2 + q3);
    __syncthreads();
    for (int s = 128; s > 0; s >>= 1) {
        if (t < s) { rs[t] += rs[t + s]; rq[t] += rq[t + s]; }
        __syncthreads();
    }
    if (t == 0) {
        const float inv_n = 1.f / (float)(SS * SS);
        float mu  = rs[0] * inv_n;
        float var = rq[0] * inv_n - mu * mu;
        mean[map]   = mu;
        invstd[map] = 1.f / sqrtf(var + 1e-5f);
    }
}

// ---------------------------------------------------------------------------
// Kernel 4: per (layer,batch): s_i = sum over heads & span of
// |(attn[l,b,h,mask_pos,p]-mu)*invstd*w[mp,p] + b[mp,p]|, plus the 8x7 anchors
// anchor[l][b][j] = mean_h attn[l,b,h,4,j].
// ---------------------------------------------------------------------------
__global__ __launch_bounds__(256)
void k_rowspan(const float* __restrict__ attn,
               const float* __restrict__ ln_w, const float* __restrict__ ln_b,
               const int* __restrict__ posw,
               const float* __restrict__ mean, const float* __restrict__ invstd,
               float* __restrict__ s_i, float* __restrict__ anchor) {
    __shared__ float red[256];
    const int lb = blockIdx.x;           // l*8+b, 0..23
    const int b  = lb & 7;
    const int t  = threadIdx.x;
    const int mp = posw[b];
    const int lo = posw[BB + b] + 1;
    const int hi = posw[2 * BB + b] - 2;

    float acc = 0.f;
    for (int idx = t; idx < HH * SS; idx += 256) {
        const int h = idx >> 9, p = idx & (SS - 1);
        const int m = lb * HH + h;
        const float x  = attn[((size_t)m * SS + mp) * SS + p];
        const float nv = fabsf((x - mean[m]) * invstd[m] * ln_w[mp * SS + p] + ln_b[mp * SS + p]);
        if (p >= lo && p <= hi) acc += nv;
    }
    red[t] = acc; __syncthreads();
    for (int s = 128; s > 0; s >>= 1) { if (t < s) red[t] += red[t + s]; __syncthreads(); }
    if (t == 0) s_i[lb] = red[0];

    if (t < 7) {
        float a = 0.f;
        for (int h = 0; h < HH; ++h)
            a += attn[(((size_t)lb * HH + h) * SS + 4) * SS + t];
        anchor[lb * 7 + t] = a * (1.f / 16.f);
    }
}

// ---------------------------------------------------------------------------
// Kernel 5: single wave32. Gram matrix G = anchor @ anchor^T per layer via
// V_WMMA_F32_16X16X4_F32 (K=7 -> two K=4 chunks, zero padded; rows 8..15 zero).
// For a Gram matrix B = A^T, and the documented 16x4-A / 4x16-B register
// layouts carry identical data, so SRC0 == SRC1. Then the serial tail:
// supcon, scan, CE, loss mix.
// ---------------------------------------------------------------------------
__global__ __launch_bounds__(32)
void k_final(const float* __restrict__ s_i, const float* __restrict__ anchor,
             const int* __restrict__ labels, float* __restrict__ out) {
    __shared__ float G[3][8][8];
    const int lane = threadIdx.x;        // 0..31, one wave, EXEC all ones
    const int r    = lane & 15;
    const bool hiH = lane >= 16;

    for (int l = 0; l < 3; ++l) {
        const float* A = anchor + l * 56;          // [8][7]
        v2f a0; a0[0] = 0.f; a0[1] = 0.f;          // K chunk 0..3
        v2f a1; a1[0] = 0.f; a1[1] = 0.f;          // K chunk 4..7 (7 = pad 0)
        if (r < 8) {
            const float* ar = A + r * 7;
            if (!hiH) { a0[0] = ar[0]; a0[1] = ar[1]; a1[0] = ar[4]; a1[1] = ar[5]; }
            else      { a0[0] = ar[2]; a0[1] = ar[3]; a1[0] = ar[6]; a1[1] = 0.f;  }
        }
        v8f acc = {0.f, 0.f, 0.f, 0.f, 0.f, 0.f, 0.f, 0.f};
        acc = __builtin_amdgcn_wmma_f32_16x16x4_f32(false, a0, false, a0,
                                                    (short)0, acc, false, false);
        acc = __builtin_amdgcn_wmma_f32_16x16x4_f32(false, a1, false, a1,
                                                    (short)0, acc, false, false);
        if (lane < 8) {
#pragma unroll
            for (int i = 0; i < 8; ++i) G[l][i][lane] = acc[i];
        }
    }
    __syncthreads();

    if (lane == 0) {
        int lab[BB];
        for (int i = 0; i < BB; ++i) lab[i] = labels[i];

        // --- supcon per layer, accumulated into gap ---
        float gap = 0.f;
        for (int l = 0; l < 3; ++l) {
            float n[BB], e[BB][BB];
            for (int i = 0; i < BB; ++i) n[i] = fmaxf(sqrtf(G[l][i][i]), 1e-8f);
            for (int i = 0; i < BB; ++i)
                for (int j = 0; j < BB; ++j) {
                    float sim = G[l][i][j] / (n[i] * n[j]);
                    e[i][j] = expf(sim * 2.0f);               // temp = 0.5
                }
            float nss[BB];
            for (int i = 0; i < BB; ++i) {
                float acc = 0.f;
                for (int j = 0; j < BB; ++j)
                    if (lab[i] != lab[j]) acc += e[i][j];      // (e - same*e).sum
                nss[i] = acc;
            }
            float total = 0.f; int cnt = 0;
            for (int i = 0; i < BB; ++i)
                for (int j = 0; j < BB; ++j) {
                    float same  = (lab[i] == lab[j]) ? 1.f : 0.f;
                    float sv    = same * e[i][j];
                    float ratio = sv / (sv + nss[i]);
                    float arg   = (1.f - same) + ratio + ((i == j) ? 1.f : 0.f);
                    float l0    = logf(arg);
                    total += l0;
                    if (l0 != 0.f) cnt++;
                }
            gap += total / (float)cnt;
        }

        // --- carried scan over (layer, batch) ---
        float tc = 0.f, ba = 0.f, layer_att = 0.f;
        for (int l = 0; l < 3; ++l) {
            for (int b = 0; b < BB; ++b) {
                tc = (tc + s_i[l * BB + b]) / 16.0f;
                ba += tc;
            }
            ba = ba / 80.0f;                 // bsz * 10
            layer_att += ba;
        }

        const float loss_1 = gap / 3.0f;
        const float loss_2 = layer_att / 3.0f;

        // --- 2-class cross entropy on out_logits (written by k_masklogits) ---
        float loss_3 = 0.f;
        for (int b = 0; b < BB; ++b) {
            float ng = out[b * 2 + 0], ps = out[b * 2 + 1];
            float mx = fmaxf(ng, ps);
            float lse = mx + logf(expf(ng - mx) + expf(ps - mx));
            float sel = (lab[b] == 0) ? ng : ps;
            loss_3 += -(sel - lse);
        }
        loss_3 /= 8.0f;

        out[16] = 0.7f * loss_3 + 0.15f * loss_1 + 0.15f * loss_2;
        out[17] = loss_1;
        out[18] = loss_2;
        out[19] = loss_3;
    }
}

// ---------------------------------------------------------------------------
extern "C" void kernel_launch(void* const* d_in, const int* in_sizes, int n_in,
                              void* d_out, int out_size, void* d_ws, size_t ws_size,
                              hipStream_t stream) {
    (void)in_sizes; (void)n_in; (void)out_size; (void)ws_size;
    const float* logits = (const float*)d_in[0];   // [8,512,50265]
    const float* attn   = (const float*)d_in[1];   // [3,8,16,512,512]
    const float* ln_w   = (const float*)d_in[2];   // [512,512]
    const float* ln_b   = (const float*)d_in[3];   // [512,512]
    const float* pos_w  = (const float*)d_in[4];   // [3]
    const float* neg_w  = (const float*)d_in[5];   // [3]
    const int*   ids    = (const int*)d_in[6];     // [8,512]
    // d_in[7] attention_mask: unused by the reference math
    const int*   labels = (const int*)d_in[8];     // [8]
    float* out = (float*)d_out;                    // 20 floats

    int*   ws_i  = (int*)d_ws;                     // 24 ints: mask_pos/left/right
    float* ws_f  = (float*)d_ws + 32;              // 128B-offset float region
    float* mean  = ws_f;                           // 384
    float* ivstd = ws_f + 384;                     // 384
    float* s_i   = ws_f + 768;                     // 24
    float* anch  = ws_f + 800;                     // 168

    k_positions <<<1,   32,  0, stream>>>(ids, ws_i);
    k_masklogits<<<BB,  256, 0, stream>>>(logits, pos_w, neg_w, ws_i, out);
    k_stats     <<<384, 256, 0, stream>>>(attn, mean, ivstd);
    k_rowspan   <<<24,  256, 0, stream>>>(attn, ln_w, ln_b, ws_i, mean, ivstd, s_i, anch);
    k_final     <<<1,   32,  0, stream>>>(s_i, anch, labels, out);
}